// PropagationOnly_SharedPixel_38079180046767
// MI455X (gfx1250) — compile-verified
//
#include <hip/hip_runtime.h>
#include <hip/hip_bf16.h>

typedef __attribute__((ext_vector_type(2))) float v2f;
typedef __attribute__((ext_vector_type(8))) float v8f;

#define HSIDE  64
#define NPIX   4096   // 64*64
#define NITER  12
#define KSTEPS 6      // 6 x K=4 covers a 24-wide window >= any tile's band

// Tridiagonal ones matrix: T[r][c] = (|r-c| <= 1)
static __device__ __forceinline__ float Tmat(int r, int c) {
    int d = r - c;
    return (d >= -1 && d <= 1) ? 1.0f : 0.0f;
}

// Start of the 24-wide K window for a 16-row/col block at offset blk.
// True band is [blk-1, blk+16]; clamped window [start, start+23] always covers it,
// and any extra K-blocks hit all-zero T fragments (exact zero contribution).
static __device__ __forceinline__ int band_start(int blk) {
    int s = blk - 4;
    if (s < 0)  s = 0;
    if (s > 40) s = 40;
    return s;
}

static __device__ __forceinline__ v8f wmma_f32(v2f a, v2f b, v8f c) {
    return __builtin_amdgcn_wmma_f32_16x16x4_f32(false, a, false, b,
                                                 (short)0, c, false, false);
}

__launch_bounds__(256, 1)
__global__ void prop_stencil_wmma_kernel(const float* __restrict__ X,
                                         const float* __restrict__ pred,
                                         const float* __restrict__ w,
                                         const float* __restrict__ a,
                                         const float* __restrict__ bias,
                                         const float* __restrict__ scalar,
                                         float* __restrict__ out)
{
    // Persistent per-sample state, resident in LDS for all 12 iterations.
    __shared__ float Ush[NPIX];   // current u (64x64 image)
    __shared__ float Psh[NPIX];   // intermediate P = T @ U
    __shared__ float Csh[NPIX];   // loop-invariant const = u_fix + bias + a*X
    __shared__ float Wsh[NPIX];   // per-pixel row scale w

    const int b   = blockIdx.x;
    const int tid = threadIdx.x;
    const float s = scalar[0];

    // ---- init: 16 pixels per thread ----
    for (int i = tid; i < NPIX; i += 256) {
        float p    = pred[b * NPIX + i];
        float ufix = (p == -1.0f) ? 0.0f : p;
        Csh[i] = ufix + bias[i] + a[i] * X[b * NPIX + i];
        Wsh[i] = w[i];
        Ush[i] = p;                          // u0 = pred
    }
    __syncthreads();

    // Wave-uniform tile coordinates, forced into SGPRs.
    const int wave  = __builtin_amdgcn_readfirstlane((int)tid >> 5);
    const int lane  = tid & 31;
    const int lrow  = lane & 15;             // m (A/D) or n (B/D) within tile
    const int lhalf = lane >> 4;             // K-pair half per ISA f32 layout

    // Stage 1 (P = T@U): pair the wave's two tiles along a row -> shared A(=T).
    const int pIb = (wave >> 1) * 16;
    const int pJ0 = ((wave * 2)     & 3) * 16;
    const int pJ1 = ((wave * 2 + 1) & 3) * 16;
    const int s1k = band_start(pIb);

    // Stage 2 (S = P@T): pair along a column -> shared B(=T).
    const int sJb = (wave >> 1) * 16;
    const int sI0 = ((wave * 2)     & 3) * 16;
    const int sI1 = ((wave * 2 + 1) & 3) * 16;
    const int s2k = band_start(sJb);

    #pragma unroll 1
    for (int it = 0; it < NITER; ++it) {
        // ---- stage 1: P = T @ U  (vertical 3-neighbor sums) ----
        {
            v8f acc0 = {}, acc1 = {};
            #pragma unroll
            for (int j = 0; j < KSTEPS; ++j) {
                const int ka = s1k + 4 * j + 2 * lhalf;
                v2f av, b0, b1;
                av.x = Tmat(pIb + lrow, ka);
                av.y = Tmat(pIb + lrow, ka + 1);
                b0.x = Ush[ka * HSIDE + pJ0 + lrow];
                b0.y = Ush[(ka + 1) * HSIDE + pJ0 + lrow];
                b1.x = Ush[ka * HSIDE + pJ1 + lrow];
                b1.y = Ush[(ka + 1) * HSIDE + pJ1 + lrow];
                acc0 = wmma_f32(av, b0, acc0);   // two independent acc chains
                acc1 = wmma_f32(av, b1, acc1);   // hide WMMA->WMMA RAW latency
            }
            #pragma unroll
            for (int v = 0; v < 8; ++v) {
                const int m = v + 8 * lhalf;     // D layout: row m, col lrow
                Psh[(pIb + m) * HSIDE + pJ0 + lrow] = acc0[v];
                Psh[(pIb + m) * HSIDE + pJ1 + lrow] = acc1[v];
            }
        }
        __syncthreads();

        // ---- stage 2: S = P @ T (horizontal sums) + fused epilogue ----
        {
            v8f acc0 = {}, acc1 = {};
            #pragma unroll
            for (int j = 0; j < KSTEPS; ++j) {
                const int ka = s2k + 4 * j + 2 * lhalf;
                v2f bv, a0, a1;
                bv.x = Tmat(ka,     sJb + lrow);
                bv.y = Tmat(ka + 1, sJb + lrow);
                a0.x = Psh[(sI0 + lrow) * HSIDE + ka];
                a0.y = Psh[(sI0 + lrow) * HSIDE + ka + 1];
                a1.x = Psh[(sI1 + lrow) * HSIDE + ka];
                a1.y = Psh[(sI1 + lrow) * HSIDE + ka + 1];
                acc0 = wmma_f32(a0, bv, acc0);
                acc1 = wmma_f32(a1, bv, acc1);
            }
            // epilogue: u = tanh(s * (const + w * S)), written back to U
            #pragma unroll
            for (int v = 0; v < 8; ++v) {
                const int m   = v + 8 * lhalf;
                const int i0  = (sI0 + m) * HSIDE + sJb + lrow;
                const int i1  = (sI1 + m) * HSIDE + sJb + lrow;
                Ush[i0] = tanhf(s * (Csh[i0] + Wsh[i0] * acc0[v]));
                Ush[i1] = tanhf(s * (Csh[i1] + Wsh[i1] * acc1[v]));
            }
        }
        __syncthreads();
    }

    // ---- write result ----
    for (int i = tid; i < NPIX; i += 256) {
        out[b * NPIX + i] = Ush[i];
    }
}

extern "C" void kernel_launch(void* const* d_in, const int* in_sizes, int n_in,
                              void* d_out, int out_size, void* d_ws, size_t ws_size,
                              hipStream_t stream) {
    // setup_inputs order: X, pred, w, a, bias, scalar, hiddenWeight, dtype
    const float* X      = (const float*)d_in[0];
    const float* pred   = (const float*)d_in[1];
    const float* w      = (const float*)d_in[2];
    const float* a      = (const float*)d_in[3];
    const float* bias   = (const float*)d_in[4];
    const float* scalar = (const float*)d_in[5];
    float* out = (float*)d_out;

    const int B = in_sizes[0] / NPIX;   // 128
    prop_stencil_wmma_kernel<<<B, 256, 0, stream>>>(X, pred, w, a, bias, scalar, out);
}